// EdgeTypeGraphConv_73409581023701
// MI455X (gfx1250) — compile-verified
//
#include <hip/hip_runtime.h>
#include <hip/hip_bf16.h>

#define N_NODES 10000
#define N_EDGES 640000
#define D 128          // D_IN == D_OUT == 128

typedef float v2f __attribute__((ext_vector_type(2)));
typedef float v8f __attribute__((ext_vector_type(8)));

// ---------------------------------------------------------------------------
// Phase 1: WMMA f32 GEMM.  grid = (625 row-tiles, 3 weight matrices).
// Block = 256 threads = 8 waves; wave w computes the 16x16 output tile at
// columns [w*16, w*16+16) for rows [blockIdx.x*16, +16).
// The 16x128 x-tile AND the full 128x128 W are staged in LDS via
// gfx1250 async-to-LDS DMA (global_load_async_to_lds_b128, ASYNCcnt).
// which==0 -> x@W0+b0 -> xt0 ; which==1 -> x@W1+b1 -> xt1 ;
// which==2 -> x@Wself+bself -> d_out (accumulator base for the scatter).
// ---------------------------------------------------------------------------
__global__ __launch_bounds__(256) void etgc_gemm_wmma(
    const float* __restrict__ x,
    const float* __restrict__ W0, const float* __restrict__ b0,
    const float* __restrict__ W1, const float* __restrict__ b1,
    const float* __restrict__ Ws, const float* __restrict__ bs,
    float* __restrict__ xt0, float* __restrict__ xt1,
    float* __restrict__ outself)
{
    extern __shared__ float smem[];        // [16*D] x-tile | [D*D] W  (72 KB)
    float* sx = smem;
    float* wl = smem + 16 * D;

    const int which = blockIdx.y;
    const float* __restrict__ W = (which == 0) ? W0 : ((which == 1) ? W1 : Ws);
    const float* __restrict__ b = (which == 0) ? b0 : ((which == 1) ? b1 : bs);
    float* __restrict__ outp     = (which == 0) ? xt0 : ((which == 1) ? xt1 : outself);

    const int nodeBase = blockIdx.x * 16;
    const int t = threadIdx.x;

    // Wave-relative LDS byte offsets (low 32 bits of the LDS-aperture address).
    const unsigned sx_lds = (unsigned)(uintptr_t)sx + (unsigned)t * 32u;
    const unsigned wl_lds = (unsigned)(uintptr_t)wl + (unsigned)t * 256u;
    const unsigned gx_off = (unsigned)t * 32u;     // 32 B of x-tile per thread
    const unsigned gw_off = (unsigned)t * 256u;    // 256 B of W per thread
    const float* gx = x + (size_t)nodeBase * D;

    // x row-tile: 8192 B total = 2 x b128 per thread (async DMA into LDS).
    asm volatile(
        "global_load_async_to_lds_b128 %0, %1, %2\n\t"
        "global_load_async_to_lds_b128 %0, %1, %2 offset:16"
        :: "v"(sx_lds), "v"(gx_off), "s"(gx) : "memory");

    // W matrix: 65536 B total = 16 x b128 per thread (async DMA into LDS).
    asm volatile(
        "global_load_async_to_lds_b128 %0, %1, %2\n\t"
        "global_load_async_to_lds_b128 %0, %1, %2 offset:16\n\t"
        "global_load_async_to_lds_b128 %0, %1, %2 offset:32\n\t"
        "global_load_async_to_lds_b128 %0, %1, %2 offset:48\n\t"
        "global_load_async_to_lds_b128 %0, %1, %2 offset:64\n\t"
        "global_load_async_to_lds_b128 %0, %1, %2 offset:80\n\t"
        "global_load_async_to_lds_b128 %0, %1, %2 offset:96\n\t"
        "global_load_async_to_lds_b128 %0, %1, %2 offset:112\n\t"
        "global_load_async_to_lds_b128 %0, %1, %2 offset:128\n\t"
        "global_load_async_to_lds_b128 %0, %1, %2 offset:144\n\t"
        "global_load_async_to_lds_b128 %0, %1, %2 offset:160\n\t"
        "global_load_async_to_lds_b128 %0, %1, %2 offset:176\n\t"
        "global_load_async_to_lds_b128 %0, %1, %2 offset:192\n\t"
        "global_load_async_to_lds_b128 %0, %1, %2 offset:208\n\t"
        "global_load_async_to_lds_b128 %0, %1, %2 offset:224\n\t"
        "global_load_async_to_lds_b128 %0, %1, %2 offset:240"
        :: "v"(wl_lds), "v"(gw_off), "s"(W) : "memory");

    // Wait for this wave's async DMA, then sync the workgroup.
    asm volatile("s_wait_asynccnt 0" ::: "memory");
    __syncthreads();

    const int lane  = threadIdx.x & 31;
    const int wave  = threadIdx.x >> 5;      // n-tile index 0..7
    const int l15   = lane & 15;
    const int hi    = lane >> 4;             // 0 => K pair {0,1}, 1 => K pair {2,3}
    const int nglob = wave * 16 + l15;       // output column for this lane

    // C/D layout (f32 16x16): lane = column, VGPR v = row (v + 8*hi).
    v8f c;
    const float bb = b[nglob];
#pragma unroll
    for (int v = 0; v < 8; ++v) c[v] = bb;

    // A fragment source: row (lane&15) of the LDS x-tile, K pair chosen by hi.
    const float* __restrict__ arow = sx + l15 * D + 2 * hi;
    // B fragment source: column nglob of LDS-resident W, rows k0+2*hi, +1.
    const float* __restrict__ bcol = wl + nglob + 2 * hi * D;

#pragma unroll
    for (int k0 = 0; k0 < D; k0 += 4) {
        v2f a = *(const v2f*)(arow + k0);    // ds_load_b64
        v2f bv;
        bv.x = bcol[k0 * D];                 // ds_load (2addr pairable)
        bv.y = bcol[k0 * D + D];
        // D = A(16x4) * B(4x16) + C   (exact f32 matrix op)
        c = __builtin_amdgcn_wmma_f32_16x16x4_f32(
                /*neg_a=*/false, a, /*neg_b=*/false, bv,
                /*c_mod=*/(short)0, c, /*reuse_a=*/false, /*reuse_b=*/false);
    }

#pragma unroll
    for (int v = 0; v < 8; ++v) {
        const int m = v + 8 * hi;
        outp[(size_t)(nodeBase + m) * D + nglob] = c[v];
    }
}

// ---------------------------------------------------------------------------
// Zero the degree accumulator.
// ---------------------------------------------------------------------------
__global__ void etgc_zero_deg(float* __restrict__ deg, int n)
{
    int i = blockIdx.x * blockDim.x + threadIdx.x;
    if (i < n) deg[i] = 0.0f;
}

// ---------------------------------------------------------------------------
// Phase 2: edge scatter.  One thread per (edge, 4-channel quad): 32 quads
// cover D=128.  float4 gather from xt[type][src], 4 atomic f32 adds into
// out[dst].  Quad 0 also accumulates the (all-edge) degree.
// ---------------------------------------------------------------------------
__global__ __launch_bounds__(256) void etgc_scatter(
    const long long* __restrict__ edge_index,   // [2, E] int64
    const long long* __restrict__ edge_types,   // [E]    int64
    const float* __restrict__ xt,               // [2, N, D]
    float* __restrict__ out,                    // [N, D] (pre-seeded with self term)
    float* __restrict__ deg)                    // [N]
{
    const long long tid = (long long)blockIdx.x * blockDim.x + threadIdx.x;
    const long long total = (long long)N_EDGES * 32;
    if (tid >= total) return;

    const int e  = (int)(tid >> 5);
    const int q  = (int)(tid & 31);
    const int c0 = q * 4;

    const int src = (int)edge_index[e];
    const int dst = (int)edge_index[(size_t)N_EDGES + e];
    const int ty  = (int)edge_types[e];

    const float4 m4 = *(const float4*)(xt + (size_t)ty * N_NODES * D
                                          + (size_t)src * D + c0);
    float* o = out + (size_t)dst * D + c0;
    atomicAdd(o + 0, m4.x);
    atomicAdd(o + 1, m4.y);
    atomicAdd(o + 2, m4.z);
    atomicAdd(o + 3, m4.w);

    if (q == 0) atomicAdd(&deg[dst], 1.0f);
}

// ---------------------------------------------------------------------------
// Phase 3: degree normalization + ReLU.
// ---------------------------------------------------------------------------
__global__ __launch_bounds__(256) void etgc_norm_relu(
    float* __restrict__ out, const float* __restrict__ deg)
{
    const int tid = blockIdx.x * blockDim.x + threadIdx.x;
    if (tid >= N_NODES * D) return;
    const int node = tid >> 7;   // /128
    float d = deg[node];
    d = (d == 0.0f) ? 1.0f : d;
    const float v = out[tid] / d;
    out[tid] = (v > 0.0f) ? v : 0.0f;
}

// ---------------------------------------------------------------------------
// Launcher.  Inputs (setup_inputs order):
//   0:x  1:edge_index  2:edge_types  3:W0  4:b0  5:W1  6:b1  7:Wself  8:bself
// Workspace: xt0 [N*D] | xt1 [N*D] | deg [N]   (~10.3 MB of f32)
// ---------------------------------------------------------------------------
extern "C" void kernel_launch(void* const* d_in, const int* in_sizes, int n_in,
                              void* d_out, int out_size, void* d_ws, size_t ws_size,
                              hipStream_t stream)
{
    const float*     x   = (const float*)d_in[0];
    const long long* ei  = (const long long*)d_in[1];
    const long long* et  = (const long long*)d_in[2];
    const float*     W0  = (const float*)d_in[3];
    const float*     b0  = (const float*)d_in[4];
    const float*     W1  = (const float*)d_in[5];
    const float*     b1  = (const float*)d_in[6];
    const float*     Ws  = (const float*)d_in[7];
    const float*     bs  = (const float*)d_in[8];
    float*           out = (float*)d_out;

    float* xt0 = (float*)d_ws;                       // [N, D]
    float* xt1 = xt0 + (size_t)N_NODES * D;          // [N, D]
    float* deg = xt1 + (size_t)N_NODES * D;          // [N]

    // Phase 0: clear degree counters.
    etgc_zero_deg<<<(N_NODES + 255) / 256, 256, 0, stream>>>(deg, N_NODES);

    // Phase 1: the three WMMA GEMMs (xt0, xt1, and self term into d_out).
    dim3 ggrid(N_NODES / 16, 3);
    const size_t smemBytes = (size_t)(16 * D + D * D) * sizeof(float);  // 73728
    etgc_gemm_wmma<<<ggrid, 256, smemBytes, stream>>>(x, W0, b0, W1, b1, Ws, bs,
                                                      xt0, xt1, out);

    // Phase 2: per-edge gather + atomic scatter-add (xt is [xt0 | xt1]).
    const long long scatterThreads = (long long)N_EDGES * 32;
    etgc_scatter<<<(unsigned)((scatterThreads + 255) / 256), 256, 0, stream>>>(
        ei, et, xt0, out, deg);

    // Phase 3: degree normalization + ReLU.
    etgc_norm_relu<<<(N_NODES * D + 255) / 256, 256, 0, stream>>>(out, deg);
}